// AM_CVRP_1D_LimitVeh_Actor_6983616824112
// MI455X (gfx1250) — compile-verified
//
#include <hip/hip_runtime.h>
#include <hip/hip_bf16.h>
#include <math.h>

// Problem constants (match reference)
#define B_   512
#define N_   100
#define E_   128
#define H_   8
#define HD_  16
#define T_   200   // 2*N
#define CLIP_ 10.0f

typedef float v2f __attribute__((ext_vector_type(2)));
typedef float v8f __attribute__((ext_vector_type(8)));

// D = A(16x4) * B(4x16) + C(16x16), all fp32. 8-arg VOP3P pattern:
// (neg_a, A, neg_b, B, c_mod, C, reuse_a, reuse_b)
__device__ __forceinline__ v8f wmma_f32_16x16x4(v2f a, v2f b, v8f c) {
  return __builtin_amdgcn_wmma_f32_16x16x4_f32(false, a, false, b, (short)0, c,
                                               false, false);
}

__device__ __forceinline__ float wave_red_add(float v) {
  for (int off = 16; off > 0; off >>= 1) v += __shfl_xor(v, off, 32);
  return v;
}
__device__ __forceinline__ float wave_red_max(float v) {
  for (int off = 16; off > 0; off >>= 1) v = fmaxf(v, __shfl_xor(v, off, 32));
  return v;
}

// ---------------------------------------------------------------------------
// Kernel 1: K/V/G projections.  OUT_p[b] = embed[b] (100x128) @ W_p (128x128)
// grid = (B, 3), block = 256 (8 waves).  Wave w owns column tile w; loops over
// 7 row tiles (100 rows padded to 112).  Real 16x16x4 f32 WMMA GEMM.
// ---------------------------------------------------------------------------
__global__ __launch_bounds__(256) void proj_kernel(
    const float* __restrict__ embed, const float* __restrict__ Wk,
    const float* __restrict__ Wv, const float* __restrict__ Wg,
    float* __restrict__ Kc, float* __restrict__ Vc, float* __restrict__ Gc) {
  const int b = blockIdx.x;
  const int p = blockIdx.y;
  const float* W = (p == 0) ? Wk : (p == 1) ? Wv : Wg;
  float* OUT = (p == 0) ? Kc : (p == 1) ? Vc : Gc;

  const int w = threadIdx.x >> 5;
  const int lane = threadIdx.x & 31;
  const int lm = lane & 15;
  const int hi2 = (lane >> 4) * 2;  // lanes 16-31 hold K+2, K+3

  const float* Eb = embed + (size_t)b * N_ * E_;
  float* Ob = OUT + (size_t)b * N_ * E_;
  const int nt = w;  // 8 waves -> 8 column tiles of 16

  for (int mt = 0; mt < 7; ++mt) {
    v8f c = {0.f, 0.f, 0.f, 0.f, 0.f, 0.f, 0.f, 0.f};
    const int m = mt * 16 + lm;       // A row for this lane
    const int col = nt * 16 + lm;     // B/C column for this lane
    const bool mOK = (m < N_);
#pragma unroll 4
    for (int k0 = 0; k0 < E_; k0 += 4) {
      const int ka = k0 + hi2;
      v2f a, bb;
      a[0] = mOK ? Eb[m * E_ + ka] : 0.f;
      a[1] = mOK ? Eb[m * E_ + ka + 1] : 0.f;
      bb[0] = W[ka * E_ + col];
      bb[1] = W[(ka + 1) * E_ + col];
      c = wmma_f32_16x16x4(a, bb, c);
    }
#pragma unroll
    for (int r = 0; r < 8; ++r) {
      const int row = mt * 16 + r + ((lane < 16) ? 0 : 8);
      if (row < N_) Ob[row * E_ + col] = c[r];
    }
  }
}

// ---------------------------------------------------------------------------
// Kernel 2: full 200-step rollout.  One workgroup per batch element, all state
// in LDS.  K/V/G stream from L2 (78.6 MB total, fits 192 MB L2).  The two
// weight GEMVs use f32 WMMA (row 0 of a 16-row tile).
// ---------------------------------------------------------------------------
__global__ __launch_bounds__(256) void decode_kernel(
    const float* __restrict__ problems, const float* __restrict__ embed,
    const float* __restrict__ avg, const int* __restrict__ vnum,
    const float* __restrict__ Wq, const float* __restrict__ Wo,
    const float* __restrict__ Kc, const float* __restrict__ Vc,
    const float* __restrict__ Gc, float* __restrict__ out) {
  __shared__ float ctx[257];     // [avg(128) | cur(128) | cap]
  __shared__ float qv[E_];
  __shared__ float att[H_][N_];  // scores, then attention weights in-place
  __shared__ float gp[E_];
  __shared__ float gl[E_];
  __shared__ float logits[N_];
  __shared__ float dem[N_];
  __shared__ int selA[N_];
  __shared__ int mskA[N_];
  __shared__ float s_cap, s_lp;
  __shared__ int s_prev, s_nxt, s_veh, s_cnt, s_done, s_rtd, s_vn;

  const int tid = threadIdx.x;
  const int w = tid >> 5;
  const int lane = tid & 31;
  const int lm = lane & 15;
  const int hi2 = (lane >> 4) * 2;
  const int b = blockIdx.x;
  const size_t bOff = (size_t)b * N_;
  const float NEG = -__builtin_inff();
  const float INV_SQRT_E = 0.08838834764831845f;  // 1/sqrt(128)

  // ----- init -----
  if (tid < E_) {
    ctx[tid] = avg[(size_t)b * E_ + tid];
    ctx[E_ + tid] = embed[bOff * E_ + tid];  // cur = graph_embed[b,0,:]
  }
  for (int n = tid; n < N_; n += 256) {
    dem[n] = problems[(bOff + n) * 3 + 2];
    selA[n] = (n == 0) ? 1 : 0;
    mskA[n] = (n == 0) ? 1 : 0;
  }
  if (tid == 0) {
    ctx[256] = 1.f;
    s_cap = 1.f; s_lp = 0.f;
    s_prev = 0; s_veh = 0; s_cnt = 1; s_done = 0; s_rtd = 0;
    s_vn = vnum[0];
    out[(size_t)b * (T_ + 2)] = 0.f;            // actions[:,0] = 0
    out[(size_t)b * (T_ + 2) + T_ + 1] = 0.f;   // actions[:,-1] = 0
  }
  __syncthreads();

#pragma unroll 1
  for (int t = 0; t < T_; ++t) {
    // ---- q = ctx @ Wq  (1x257 @ 257x128) via WMMA matvec; wave w -> cols ----
    {
      v8f c = {0.f, 0.f, 0.f, 0.f, 0.f, 0.f, 0.f, 0.f};
      const int col = w * 16 + lm;
#pragma unroll 4
      for (int k0 = 0; k0 < 256; k0 += 4) {
        const int ka = k0 + hi2;
        v2f a, bb;
        a[0] = (lm == 0) ? ctx[ka] : 0.f;       // only M=0 row populated
        a[1] = (lm == 0) ? ctx[ka + 1] : 0.f;
        bb[0] = Wq[ka * E_ + col];
        bb[1] = Wq[(ka + 1) * E_ + col];
        c = wmma_f32_16x16x4(a, bb, c);
      }
      if (lane < 16)  // C row 0 lives in VGPR0, lanes 0..15
        qv[col] = c[0] + ctx[256] * Wq[256 * E_ + col];  // k=256 remainder
    }
    __syncthreads();

    // ---- scores: sc[h,n] = q[h].K[n,h]/4, masked -> -inf ----
    for (int idx = tid; idx < H_ * N_; idx += 256) {
      const int n = idx % N_, h = idx / N_;
      const float* kp = Kc + ((bOff + n) * H_ + h) * HD_;
      const float* qp = qv + h * HD_;
      float d = 0.f;
#pragma unroll
      for (int i = 0; i < HD_; ++i) d += qp[i] * kp[i];
      att[h][n] = mskA[n] ? NEG : d * 0.25f;
    }
    __syncthreads();

    // ---- softmax over n, head = wave id ----
    {
      float m = NEG;
      for (int n = lane; n < N_; n += 32) m = fmaxf(m, att[w][n]);
      m = wave_red_max(m);
      float s = 0.f;
      for (int n = lane; n < N_; n += 32) {
        const float e = (m == NEG) ? 0.f : expf(att[w][n] - m);
        att[w][n] = e;
        s += e;
      }
      s = wave_red_add(s);
      const float inv = (s > 0.f) ? 1.f / s : 0.f;
      for (int n = lane; n < N_; n += 32) att[w][n] *= inv;
    }
    __syncthreads();

    // ---- glimpse pre-proj: gp[h*16+d] = sum_n att[h,n] * V[n,h,d] ----
    if (tid < E_) {
      const int h = tid >> 4, d = tid & 15;
      float acc = 0.f;
      for (int n = 0; n < N_; ++n)
        acc += att[h][n] * Vc[((bOff + n) * H_ + h) * HD_ + d];
      gp[tid] = acc;
    }
    __syncthreads();

    // ---- glimpse = gp @ Wo  (1x128 @ 128x128) via WMMA matvec ----
    {
      v8f c = {0.f, 0.f, 0.f, 0.f, 0.f, 0.f, 0.f, 0.f};
      const int col = w * 16 + lm;
#pragma unroll 4
      for (int k0 = 0; k0 < E_; k0 += 4) {
        const int ka = k0 + hi2;
        v2f a, bb;
        a[0] = (lm == 0) ? gp[ka] : 0.f;
        a[1] = (lm == 0) ? gp[ka + 1] : 0.f;
        bb[0] = Wo[ka * E_ + col];
        bb[1] = Wo[(ka + 1) * E_ + col];
        c = wmma_f32_16x16x4(a, bb, c);
      }
      if (lane < 16) gl[col] = c[0];
    }
    __syncthreads();

    // ---- logits[n] = CLIP*tanh(glimpse.G[n]/sqrt(E)), masked -> -inf ----
    for (int n = w; n < N_; n += 8) {
      const float4* g4 = (const float4*)(Gc + (bOff + n) * E_);
      const float4 g = g4[lane];
      float d = g.x * gl[lane * 4] + g.y * gl[lane * 4 + 1] +
                g.z * gl[lane * 4 + 2] + g.w * gl[lane * 4 + 3];
      d = wave_red_add(d);
      if (lane == 0) {
        const float v = CLIP_ * tanhf(d * INV_SQRT_E);
        logits[n] = mskA[n] ? NEG : v;
      }
    }
    __syncthreads();

    // ---- argmax + log-prob of pick: logp[argmax] = -log(sum exp(l - max)) ----
    if (w == 0) {
      float bv = NEG;
      int bi = 0;
      for (int n = lane; n < N_; n += 32) {
        const float v = logits[n];
        if (v > bv) { bv = v; bi = n; }
      }
      for (int off = 16; off > 0; off >>= 1) {
        const float ov = __shfl_xor(bv, off, 32);
        const int oi = __shfl_xor(bi, off, 32);
        if (ov > bv || (ov == bv && oi < bi)) { bv = ov; bi = oi; }
      }
      float s = 0.f;
      if (bv != NEG) {
        for (int n = lane; n < N_; n += 32) s += expf(logits[n] - bv);
        s = wave_red_add(s);
      }
      if (lane == 0) {
        s_nxt = bi;
        if (bv != NEG) s_lp += -logf(s);
      }
    }
    __syncthreads();

    // ---- scalar state update (reference order) ----
    if (tid == 0) {
      const int nxt = s_nxt;
      if (s_prev == 0 && nxt != 0) s_veh += 1;
      if (nxt != 0) s_cnt += 1;
      s_cap = (nxt == 0) ? 1.f : s_cap - dem[nxt];
      selA[nxt] = 1;
      s_rtd = (nxt == 0) ? 1 : 0;
      s_done = (s_cnt == N_) ? 1 : 0;
      s_prev = nxt;
      ctx[256] = s_cap;
      out[(size_t)b * (T_ + 2) + 1 + t] = (float)nxt;
    }
    __syncthreads();

    // ---- cur + mask rebuild ----
    {
      const int nxt = s_nxt;
      const float cap = s_cap;
      const int done = s_done;
      if (tid < E_) ctx[E_ + tid] = embed[(bOff + nxt) * E_ + tid];
      for (int n = tid; n < N_; n += 256)
        if (n > 0) mskA[n] = done ? 1 : ((selA[n] || (cap < dem[n])) ? 1 : 0);
      if (w == 0) {
        float rest = 0.f;
        int alld = 1;
        for (int n = lane; n < N_; n += 32) {
          if (!selA[n]) rest += dem[n];
          if (n >= 1 && !(cap < dem[n])) alld = 0;
        }
        rest = wave_red_add(rest);
        for (int off = 16; off > 0; off >>= 1) alld &= __shfl_xor(alld, off, 32);
        if (lane == 0) {
          const float orc = fmaxf((float)(s_vn - s_veh), 0.f);
          const int skw = (!s_rtd) && (!alld) && (orc < rest);
          mskA[0] = done ? 0 : ((s_rtd || skw) ? 1 : 0);
        }
      }
    }
    __syncthreads();
  }

  if (tid == 0) {
    const size_t ACT = (size_t)B_ * (T_ + 2);
    out[ACT + b] = s_lp;                               // log_probs
    out[ACT + B_ + b] = (s_veh > s_vn) ? 1.f : 0.f;    // infeasible
  }
}

// ---------------------------------------------------------------------------
extern "C" void kernel_launch(void* const* d_in, const int* in_sizes, int n_in,
                              void* d_out, int out_size, void* d_ws,
                              size_t ws_size, hipStream_t stream) {
  (void)in_sizes; (void)n_in; (void)out_size; (void)ws_size;
  const float* problems = (const float*)d_in[0];  // (B,N,3)
  const float* embed    = (const float*)d_in[1];  // (B,N,E)
  const float* avg      = (const float*)d_in[2];  // (B,E)
  const int*   vnum     = (const int*)d_in[3];    // scalar 25
  const float* Wk       = (const float*)d_in[4];  // (E,E)
  const float* Wv       = (const float*)d_in[5];  // (E,E)
  const float* Wg       = (const float*)d_in[6];  // (E,E)
  const float* Wq       = (const float*)d_in[7];  // (2E+1,E)
  const float* Wo       = (const float*)d_in[8];  // (E,E)
  float* out = (float*)d_out;

  // Workspace: K, V, G each B*N*E fp32 -> 78.6 MB total (fits L2).
  float* Kc = (float*)d_ws;
  float* Vc = Kc + (size_t)B_ * N_ * E_;
  float* Gc = Vc + (size_t)B_ * N_ * E_;

  proj_kernel<<<dim3(B_, 3), 256, 0, stream>>>(embed, Wk, Wv, Wg, Kc, Vc, Gc);
  decode_kernel<<<B_, 256, 0, stream>>>(problems, embed, avg, vnum, Wq, Wo, Kc,
                                        Vc, Gc, out);
}